// GMM_81724637708557
// MI455X (gfx1250) — compile-verified
//
#include <hip/hip_runtime.h>
#include <math.h>

// Problem constants
#define KK 8
#define RR 64
#define DD 256
#define BB 256
#define MOMF 0.9f
#define OMMF 0.1f
#define LOG_PI 1.1447298858494002f

typedef __attribute__((ext_vector_type(2))) float v2f;
typedef __attribute__((ext_vector_type(8))) float v8f;
typedef __attribute__((ext_vector_type(4))) int   v4i;

// ---- d_out layout (tuple return order, flat float32) ----
// out, S1(1,K,R), S2(1,K,R,D), S3(1,K,R,D,D), mu_up(1,K,R,D), phi(1,K,R),
// cov_new(1,K,R,D,D), aff(B,K,R)
#define OFF_OUT ((size_t)0)
#define OFF_S1  ((size_t)1)
#define OFF_S2  ((size_t)513)
#define OFF_S3  ((size_t)131585)
#define OFF_MU  ((size_t)33686017)
#define OFF_PHI ((size_t)33817089)
#define OFF_COV ((size_t)33817601)
#define OFF_AFF ((size_t)67372033)

// ---- workspace layout (floats) ----
#define WS_INV     ((size_t)0)        // K*R*D
#define WS_LOGDET  ((size_t)131072)   // K*R
#define WS_AFFT    ((size_t)131584)   // K*R*B  (aff transposed: [k][r][b])
#define WS_SIG     ((size_t)262656)   // K*R*D  sigma diag
#define WS_INV2    ((size_t)393728)   // K*R*D
#define WS_MI      ((size_t)524800)   // K*R*D  mu_up*inv2
#define WS_LOGDET2 ((size_t)655872)   // K*R
#define WS_T3B     ((size_t)656384)   // K*R
#define WS_GM      ((size_t)656896)   // K*R
#define WS_SX      ((size_t)657408)   // B*D
#define WS_SXX     ((size_t)722944)   // B*D
#define WS_D1      ((size_t)788480)   // B*R
#define WS_ROW     ((size_t)804864)   // B*R
// total 821248 floats = ~3.3 MB

#define AS1 __attribute__((address_space(1)))
#define AS3 __attribute__((address_space(3)))

#if defined(__has_builtin)
#if __has_builtin(__builtin_amdgcn_global_load_async_to_lds_b128) && \
    __has_builtin(__builtin_amdgcn_global_load_async_to_lds_b32)
#define HAVE_ASYNC_LDS 1
#endif
#endif
#ifndef HAVE_ASYNC_LDS
#define HAVE_ASYNC_LDS 0
#endif

#if HAVE_ASYNC_LDS
__device__ __forceinline__ void async_b128(const void* g, void* l) {
    __builtin_amdgcn_global_load_async_to_lds_b128(
        (AS1 v4i*)(uintptr_t)g,
        (AS3 v4i*)(uint32_t)(uintptr_t)l, 0, 0);
}
__device__ __forceinline__ void async_b32(const void* g, void* l) {
    __builtin_amdgcn_global_load_async_to_lds_b32(
        (AS1 int*)(uintptr_t)g,
        (AS3 int*)(uint32_t)(uintptr_t)l, 0, 0);
}
#define WAIT_ASYNC_16() asm volatile("s_wait_asynccnt 0x10" ::: "memory")
#define WAIT_ASYNC_0()  asm volatile("s_wait_asynccnt 0x0" ::: "memory")
#endif

__device__ __forceinline__ float blockReduce256(float v, float* red) {
    int t = threadIdx.x;
    red[t] = v;
    __syncthreads();
    for (int s = 128; s > 0; s >>= 1) {
        if (t < s) red[t] += red[t + s];
        __syncthreads();
    }
    float r = red[0];
    __syncthreads();
    return r;
}

// K0: inverse covariance diagonal + logdet per (k,r)
__global__ void k_prep(const float* __restrict__ cov, float* __restrict__ ws) {
    __shared__ float red[256];
    int kr = blockIdx.x, t = threadIdx.x;
    float c = cov[((size_t)kr * DD + t) * DD + t];
    float ld = blockReduce256(logf(fabsf(c)), red);
    ws[WS_INV + (size_t)kr * DD + t] = 1.0f / c;
    if (t == 0) ws[WS_LOGDET + kr] = ld;
}

// K1: responsibilities aff[b,k,r]
__global__ void k_aff(const float* __restrict__ in, const float* __restrict__ mu,
                      const float* __restrict__ pis, float* __restrict__ ws,
                      float* __restrict__ out) {
    __shared__ float red[256];
    __shared__ float maha[KK];
    __shared__ float wn[KK];
    int br = blockIdx.x;
    int b = br % BB, r = br / BB, t = threadIdx.x;
    float xv = in[(size_t)(r * BB + b) * DD + t];
    for (int k = 0; k < KK; ++k) {
        int kr = k * RR + r;
        float m  = mu[(size_t)kr * DD + t];
        float iv = ws[WS_INV + (size_t)kr * DD + t];
        float d  = xv - m;
        float s  = blockReduce256(d * d * iv, red);
        if (t == 0) maha[k] = s;
    }
    __syncthreads();
    if (t < KK) {
        int kr = t * RR + r;
        float logdens = -0.5f * (ws[WS_LOGDET + kr] + maha[t] + (float)DD * LOG_PI);
        wn[t] = pis[kr] * logdens;
    }
    __syncthreads();
    if (t < KK) {
        float denom = 0.f;
        for (int k = 0; k < KK; ++k) denom += wn[k];
        float a = wn[t] / denom;
        out[OFF_AFF + (size_t)b * KK * RR + t * RR + r] = a;
        ws[WS_AFFT + (size_t)(t * RR + r) * BB + b] = a;
    }
}

// K2: S1, gm, S2
__global__ void k_stats(const float* __restrict__ in, const float* __restrict__ S1_old,
                        const float* __restrict__ S2_old, float* __restrict__ ws,
                        float* __restrict__ out) {
    __shared__ float red[256];
    int kr = blockIdx.x, r = kr % RR, t = threadIdx.x;
    float a = ws[WS_AFFT + (size_t)kr * BB + t];
    float suma = blockReduce256(a, red);
    if (t == 0) {
        out[OFF_S1 + kr] = MOMF * S1_old[kr] + OMMF * suma;
        ws[WS_GM + kr] = suma / (float)BB;
    }
    float acc = 0.f;
    for (int b = 0; b < BB; ++b) {
        float ab = ws[WS_AFFT + (size_t)kr * BB + b];
        acc += ab * in[(size_t)(r * BB + b) * DD + t];
    }
    out[OFF_S2 + (size_t)kr * DD + t] =
        MOMF * S2_old[(size_t)kr * DD + t] + OMMF * acc;
}

// K3: phi and normalized mu_up
__global__ void k_phimu(float* __restrict__ ws, float* __restrict__ out) {
    __shared__ float red[256];
    int kr = blockIdx.x, r = kr % RR, t = threadIdx.x;
    float s1 = out[OFF_S1 + kr];
    if (t == 0) {
        float ssum = 0.f;
        for (int k2 = 0; k2 < KK; ++k2) ssum += out[OFF_S1 + k2 * RR + r];
        out[OFF_PHI + kr] = s1 / ssum;
    }
    float m = out[OFF_S2 + (size_t)kr * DD + t] / s1;
    float n2 = blockReduce256(m * m, red);
    float nrm = fmaxf(sqrtf(n2), 1e-12f);
    out[OFF_MU + (size_t)kr * DD + t] = m / nrm;
}

// K4: S3 = MOM*S3_old + 0.1 * X^T diag(aff) X via f32 WMMA 16x16x4,
//     async double-buffered LDS staging, fused cov_new + sigma-diag epilogue.
// Grid: (k,r,strip) = K*R*(D/32). Block: 256 thr = 8 waves; wave owns 4 tiles
// in one tile-row (tr = wave>>2), so the A fragment is shared across tiles.
__global__ void __launch_bounds__(256)
k_s3(const float* __restrict__ in, const float* __restrict__ S3_old,
     float* __restrict__ ws, float* __restrict__ out) {
    extern __shared__ float lds[];          // buf0[16384] buf1[16384] affs[256]
    float* buf0 = lds;
    float* buf1 = lds + 16384;
    float* affs = lds + 32768;
    int blk = blockIdx.x;
    int strip = blk & 7;
    int kr = blk >> 3;
    int r = kr % RR;
    int t = threadIdx.x;
    int wave = t >> 5, lane = t & 31;
    int half = lane >> 4, l = lane & 15;
    int d0 = strip * 32;
    int tr = wave >> 2;                     // tile row (shared by wave's 4 tiles)
    int dtl = d0 + tr * 16 + l;             // A-fragment column (output row)

    const float* xbase = in + (size_t)(r * BB) * DD;          // 256 KB slab
    const float* asrc  = ws + WS_AFFT + (size_t)kr * BB;      // 256 aff values

    v8f acc[4] = {};

#if HAVE_ASYNC_LDS
    // prologue: all aff (covers 4 chunks) + chunk 0. Per-wave ASYNCcnt = 17.
    async_b32(asrc + t, affs + t);
    #pragma unroll
    for (int i = 0; i < 16; ++i)
        async_b128(xbase + i * 1024 + t * 4, buf0 + i * 1024 + t * 4);
#else
    affs[t] = asrc[t];
    #pragma unroll
    for (int i = 0; i < 64; ++i)
        buf0[i * 256 + t] = xbase[(size_t)i * DD + t];
#endif

    for (int bc = 0; bc < 4; ++bc) {        // 4 chunks of 64 samples
        float* xs = (bc & 1) ? buf1 : buf0;
#if HAVE_ASYNC_LDS
        if (bc < 3) {                       // prefetch next chunk into other buffer
            float* nb = (bc & 1) ? buf0 : buf1;
            const float* ng = xbase + (size_t)(bc + 1) * 16384;
            #pragma unroll
            for (int i = 0; i < 16; ++i)
                async_b128(ng + i * 1024 + t * 4, nb + i * 1024 + t * 4);
            WAIT_ASYNC_16();                // newest 16 = next chunk; current done
        } else {
            WAIT_ASYNC_0();
        }
        __syncthreads();
#else
        if (bc > 0) {
            const float* cg = xbase + (size_t)bc * 16384;
            #pragma unroll
            for (int i = 0; i < 64; ++i)
                xs[i * 256 + t] = cg[(size_t)i * DD + t];
        }
        __syncthreads();
#endif
        const float* afc = affs + bc * 64;
        #pragma unroll
        for (int bb = 0; bb < 16; ++bb) {
            int b0 = bb * 4 + 2 * half;     // A layout: v0=K{0|2}, v1=K{1|3} per half
            v2f a;
            a.x = xs[(b0 + 0) * 256 + dtl];
            a.y = xs[(b0 + 1) * 256 + dtl];
            float w0 = afc[b0 + 0], w1 = afc[b0 + 1];
            #pragma unroll
            for (int j = 0; j < 4; ++j) {
                int et = ((wave & 3) * 4 + j) * 16 + l;
                v2f bv;
                bv.x = w0 * xs[(b0 + 0) * 256 + et];
                bv.y = w1 * xs[(b0 + 1) * 256 + et];
                acc[j] = __builtin_amdgcn_wmma_f32_16x16x4_f32(
                    false, a, false, bv, (short)0, acc[j], false, false);
            }
        }
        __syncthreads();                    // done reading xs before overwrite
    }

    float s1 = out[OFF_S1 + kr];
    #pragma unroll
    for (int j = 0; j < 4; ++j) {
        int tc = (wave & 3) * 4 + j;
        #pragma unroll
        for (int vj = 0; vj < 8; ++vj) {
            int drow = d0 + tr * 16 + vj + half * 8;  // C/D: VGPR vj = rows {vj, vj+8}
            int ecol = tc * 16 + l;
            size_t idx = ((size_t)kr * DD + drow) * DD + ecol;
            float g = acc[j][vj];
            float s3 = MOMF * S3_old[idx] + OMMF * g;
            out[OFF_S3 + idx] = s3;
            float cv = 0.f;
            if (drow == ecol) {
                float s2d = out[OFF_S2 + (size_t)kr * DD + drow];
                float mval = s2d / s1;
                cv = s3 / s1 - mval * mval;
                ws[WS_SIG + (size_t)kr * DD + drow] = cv;
            }
            out[OFF_COV + idx] = cv;
        }
    }
}

// K5a: Sx, Sxx per (b,d)
__global__ void k_sxx(const float* __restrict__ in, float* __restrict__ ws) {
    int b = blockIdx.x, t = threadIdx.x;
    float sx = 0.f, sxx = 0.f;
    for (int r = 0; r < RR; ++r) {
        float v = in[(size_t)(r * BB + b) * DD + t];
        sx += v; sxx += v * v;
    }
    ws[WS_SX + (size_t)b * DD + t] = sx;
    ws[WS_SXX + (size_t)b * DD + t] = sxx;
}

// K5b: inv2, mi, logdet2, t3b
__global__ void k_prep2(float* __restrict__ ws, const float* __restrict__ out) {
    __shared__ float red[256];
    int kr = blockIdx.x, t = threadIdx.x;
    float sd = ws[WS_SIG + (size_t)kr * DD + t];
    float iv = 1.0f / sd;
    float m = out[OFF_MU + (size_t)kr * DD + t];
    ws[WS_INV2 + (size_t)kr * DD + t] = iv;
    ws[WS_MI + (size_t)kr * DD + t] = m * iv;
    float ld = blockReduce256(logf(fabsf(sd)), red);
    float t3 = blockReduce256(m * m * iv, red);
    if (t == 0) { ws[WS_LOGDET2 + kr] = ld; ws[WS_T3B + kr] = t3; }
}

// K5c: per-(b,r) weighted log-likelihood pieces (diag term and s-row-sum)
__global__ void k_ll(const float* __restrict__ in, float* __restrict__ ws,
                     const float* __restrict__ out) {
    __shared__ float red[256];
    __shared__ float qs[KK], p2s[KK], p3s[KK];
    int br = blockIdx.x;
    int b = br % BB, r = br / BB, t = threadIdx.x;
    float xv   = in[(size_t)(r * BB + b) * DD + t];
    float sxv  = ws[WS_SX  + (size_t)b * DD + t];
    float sxxv = ws[WS_SXX + (size_t)b * DD + t];
    for (int k = 0; k < KK; ++k) {
        int kr = k * RR + r;
        float m   = out[OFF_MU + (size_t)kr * DD + t];
        float iv  = ws[WS_INV2 + (size_t)kr * DD + t];
        float miv = ws[WS_MI   + (size_t)kr * DD + t];
        float dd  = xv - m;
        float q  = blockReduce256(dd * dd * iv, red);
        float p2 = blockReduce256(sxxv * iv, red);
        float p3 = blockReduce256(sxv * miv, red);
        if (t == 0) { qs[k] = q; p2s[k] = p2; p3s[k] = p3; }
    }
    __syncthreads();
    if (t == 0) {
        float d1c = 0.f, rowc = 0.f;
        for (int k = 0; k < KK; ++k) {
            int kr = k * RR + r;
            float lphi = logf(out[OFF_PHI + kr]);
            float ld2  = ws[WS_LOGDET2 + kr];
            float a    = ws[WS_AFFT + (size_t)kr * BB + b];
            float logp_diag = lphi - 0.5f * (ld2 + qs[k] + (float)DD * LOG_PI);
            float sum_diff  = p2s[k] - 2.f * p3s[k] + (float)RR * ws[WS_T3B + kr];
            float sum_s = (float)RR * lphi
                        - 0.5f * ((float)RR * ld2 + sum_diff + (float)(RR * DD) * LOG_PI);
            d1c  += a * logp_diag;
            rowc += a * sum_s;
        }
        ws[WS_D1  + (size_t)b * RR + r] = d1c;
        ws[WS_ROW + (size_t)b * RR + r] = rowc - 1.0f;   // minus eye row-sum
    }
}

// K5d: final scalar
__global__ void k_final(float* __restrict__ ws, float* __restrict__ out) {
    __shared__ float red[64];
    int t = threadIdx.x;   // 64 threads, t == r
    float d1 = 0.f, o3 = 0.f;
    for (int b = 0; b < BB; ++b) {
        d1 += ws[WS_D1  + (size_t)b * RR + t];
        o3 += ws[WS_ROW + (size_t)b * RR + t];
    }
    d1 /= (float)BB; o3 /= (float)BB;
    float out2 = 0.f;
    for (int k = 0; k < KK; ++k) {
        float g = ws[WS_GM + k * RR + t];
        out2 += logf(g) * g;
    }
    red[t] = -d1 + out2 + o3;
    __syncthreads();
    for (int s = 32; s > 0; s >>= 1) {
        if (t < s) red[t] += red[t + s];
        __syncthreads();
    }
    if (t == 0) out[OFF_OUT] = red[0];
}

extern "C" void kernel_launch(void* const* d_in, const int* in_sizes, int n_in,
                              void* d_out, int out_size, void* d_ws, size_t ws_size,
                              hipStream_t stream) {
    const float* in     = (const float*)d_in[0];
    const float* mu     = (const float*)d_in[1];
    const float* pis    = (const float*)d_in[2];
    const float* cov    = (const float*)d_in[3];
    const float* S1_old = (const float*)d_in[4];
    const float* S2_old = (const float*)d_in[5];
    const float* S3_old = (const float*)d_in[6];
    float* out = (float*)d_out;
    float* ws  = (float*)d_ws;

    size_t lds_bytes = (2 * 64 * 256 + 256) * sizeof(float);  // 129 KB

    k_prep  <<<KK * RR,     256, 0, stream>>>(cov, ws);
    k_aff   <<<BB * RR,     256, 0, stream>>>(in, mu, pis, ws, out);
    k_stats <<<KK * RR,     256, 0, stream>>>(in, S1_old, S2_old, ws, out);
    k_phimu <<<KK * RR,     256, 0, stream>>>(ws, out);
    k_s3    <<<KK * RR * 8, 256, lds_bytes, stream>>>(in, S3_old, ws, out);
    k_sxx   <<<BB,          256, 0, stream>>>(in, ws);
    k_prep2 <<<KK * RR,     256, 0, stream>>>(ws, out);
    k_ll    <<<BB * RR,     256, 0, stream>>>(in, ws, out);
    k_final <<<1,            64, 0, stream>>>(ws, out);
}